// Custom_Attention_1838246002758
// MI455X (gfx1250) — compile-verified
//
#include <hip/hip_runtime.h>

typedef float v2f __attribute__((ext_vector_type(2)));
typedef float v8f __attribute__((ext_vector_type(8)));

#define N_KEYS 4096
#define BATCH 16
#define THREADS 256
#define WAVES 8
#define ROWS_PER_WAVE 16
#define ROWS_PER_BLOCK 128
#define BLOCKS_PER_BATCH (N_KEYS / ROWS_PER_BLOCK)   // 32

// out[b,i] = sum_j exp2(x_j * s_i * C - m2_i) * x_j / sum_j exp2(x_j * s_i * C - m2_i)
// with C = log2(e)/16 folded into the B operand, and m2_i a per-row shift for stability.
__global__ __launch_bounds__(THREADS)
void attn_rank1_wmma(const float* __restrict__ x, float* __restrict__ out) {
  __shared__ float xs[N_KEYS];
  __shared__ float wred[2 * WAVES];

  const int b        = blockIdx.x / BLOCKS_PER_BATCH;
  const int rowBlock = blockIdx.x % BLOCKS_PER_BATCH;
  const int tid  = threadIdx.x;
  const int lane = tid & 31;
  const int wave = tid >> 5;

  const float* xb = x + b * N_KEYS;

  // ---- Stage x[b, :] into LDS (float4) and fold per-thread min/max ----
  const float4* xb4 = (const float4*)xb;
  float4* xs4 = (float4*)xs;
  float lmax = -1e30f, lmin = 1e30f;
  for (int t = tid; t < N_KEYS / 4; t += THREADS) {
    float4 v = xb4[t];
    xs4[t] = v;
    lmax = fmaxf(fmaxf(lmax, v.x), fmaxf(v.y, fmaxf(v.z, v.w)));
    lmin = fminf(fminf(lmin, v.x), fminf(v.y, fminf(v.z, v.w)));
  }
  // wave32 reduction
  #pragma unroll
  for (int off = 16; off >= 1; off >>= 1) {
    lmax = fmaxf(lmax, __shfl_xor(lmax, off, 32));
    lmin = fminf(lmin, __shfl_xor(lmin, off, 32));
  }
  if (lane == 0) { wred[wave] = lmax; wred[WAVES + wave] = lmin; }
  __syncthreads();
  float maxx = wred[0], minx = wred[WAVES];
  #pragma unroll
  for (int w = 1; w < WAVES; ++w) {
    maxx = fmaxf(maxx, wred[w]);
    minx = fminf(minx, wred[WAVES + w]);
  }

  // ---- Per-wave query block setup ----
  const int   ibase = rowBlock * ROWS_PER_BLOCK + wave * ROWS_PER_WAVE;
  const int   col   = lane & 15;              // column i of the C/D tile this lane owns
  const int   hi    = (lane >> 4) << 3;       // 0 for lanes 0-15 (rows j=0..7), 8 for lanes 16-31
  const float C     = 1.4426950408889634f / 16.0f;   // log2(e)/16, constant-folded

  const float sc = xs[ibase + col] * C;       // per-column slope in log2 domain
  // B operand (4x16, only K=0 row nonzero): VGPR0 lanes0-15 = s_i*C, rest 0
  v2f Bm; Bm.x = (lane < 16) ? sc : 0.0f; Bm.y = 0.0f;
  // row-max of log2-scores for column i (rank-1 => from batch min/max)
  const float m2 = fmaxf(sc * maxx, sc * minx);

  float num = 0.0f, den = 0.0f;

  // ---- Sweep all 4096 keys in 16-wide tiles ----
  for (int jt = 0; jt < N_KEYS; jt += 16) {
    // A operand (16x4, only K=0 column nonzero): lane m holds x[jt+m] in VGPR0
    float xa = xs[jt + col];
    v2f Am; Am.x = (lane < 16) ? xa : 0.0f; Am.y = 0.0f;

    v8f acc = {};
    // S2'[j][i] = x_{jt+j} * s_i * C  (transposed score tile, log2 domain)
    acc = __builtin_amdgcn_wmma_f32_16x16x4_f32(
        /*neg_a=*/false, Am, /*neg_b=*/false, Bm,
        /*c_mod=*/(short)0, acc, /*reuse_a=*/false, /*reuse_b=*/false);

    // C/D layout: VGPR v holds row j = v + hi, column i = lane&15.
    // Reduce over the 8 rows this lane-half covers.
    #pragma unroll
    for (int v = 0; v < 8; ++v) {
      float e = __builtin_amdgcn_exp2f(acc[v] - m2);
      float w = xs[jt + hi + v];     // value x_j for this row (LDS broadcast per half)
      num = fmaf(e, w, num);
      den += e;
    }
  }

  // Each column i accumulated half the j's in lane and half in lane^16: merge.
  num += __shfl_xor(num, 16, 32);
  den += __shfl_xor(den, 16, 32);

  if (lane < 16) out[b * N_KEYS + ibase + lane] = num / den;
}

extern "C" void kernel_launch(void* const* d_in, const int* in_sizes, int n_in,
                              void* d_out, int out_size, void* d_ws, size_t ws_size,
                              hipStream_t stream) {
  const float* x = (const float*)d_in[0];
  float* out = (float*)d_out;
  dim3 grid(BATCH * BLOCKS_PER_BATCH);  // 512 blocks, 8 wave32 each
  attn_rank1_wmma<<<grid, THREADS, 0, stream>>>(x, out);
}